// Pooling_58119497450174
// MI455X (gfx1250) — compile-verified
//
#include <hip/hip_runtime.h>
#include <hip/hip_bf16.h>

typedef __attribute__((ext_vector_type(2))) float v2f;
typedef __attribute__((ext_vector_type(8))) float v8f;

#define N_NODES 100000
#define N_EDGES 1600000
#define DIN     128
#define DBLOOM  16
#define DOUT    128
#define DINTER  144   // DBLOOM + DOUT
#define PF      4
#define C_CLUST (N_NODES / PF)   // 25000
#define SORT_N  (1 << 21)        // 2097152 >= N_EDGES
#define SENTINEL 0x7fffffff

// ---------------------------------------------------------------- zero fill
__global__ void zero_f32_kernel(float* __restrict__ p, int n) {
    int i = blockIdx.x * blockDim.x + threadIdx.x;
    if (i < n) p[i] = 0.0f;
}

// ---------------------------------------------------------------- edge msg + scatter
// One wave (32 lanes) handles a tile of 16 edges. Computes
// msg = x[src] @ W_conv (via V_WMMA_F32_16X16X4_F32) + edge_attr @ W_edge,
// then atomically scatters into out[dst].
__global__ void edge_msg_kernel(const float* __restrict__ x,
                                const int*   __restrict__ ei,
                                const float* __restrict__ ea,
                                const float* __restrict__ Wc,
                                const float* __restrict__ We,
                                float*       __restrict__ out) {
    int wave = (blockIdx.x * blockDim.x + threadIdx.x) >> 5;
    int lane = threadIdx.x & 31;
    int row  = lane & 15;       // A-matrix row / B,C,D column-within-tile
    int half = lane >> 4;       // 0: K={0,1} / rows 0-7 ; 1: K={2,3} / rows 8-15
    int e0   = wave * 16;

    int s = ei[e0 + row];                       // src index for this lane's A row
    const float* xr = x + (size_t)s * DIN;

    v8f acc[9] = {};                            // 9 N-tiles of 16 (144 cols)

#pragma unroll 1
    for (int kb = 0; kb < DIN; kb += 4) {
        int k = kb + 2 * half;
        v2f a; a.x = xr[k]; a.y = xr[k + 1];
#pragma unroll
        for (int j = 0; j < 9; ++j) {
            int col = j * 16 + row;
            v2f b; b.x = Wc[(size_t)k * DINTER + col];
                   b.y = Wc[(size_t)(k + 1) * DINTER + col];
            acc[j] = __builtin_amdgcn_wmma_f32_16x16x4_f32(
                false, a, false, b, (short)0, acc[j], false, false);
        }
    }

    // per-row edge_attr and dst for the 8 C/D rows this lane holds
    float ea0[8], ea1[8], ea2[8]; int drow[8];
#pragma unroll
    for (int v = 0; v < 8; ++v) {
        int e = e0 + v + 8 * half;
        ea0[v] = ea[(size_t)e * 3 + 0];
        ea1[v] = ea[(size_t)e * 3 + 1];
        ea2[v] = ea[(size_t)e * 3 + 2];
        drow[v] = ei[N_EDGES + e];
    }

#pragma unroll
    for (int j = 0; j < 9; ++j) {
        int col = j * 16 + row;
        float w0 = We[col], w1 = We[DINTER + col], w2 = We[2 * DINTER + col];
#pragma unroll
        for (int v = 0; v < 8; ++v) {
            float val = acc[j][v] + ea0[v] * w0 + ea1[v] * w1 + ea2[v] * w2;
            atomicAdd(&out[(size_t)drow[v] * DINTER + col], val);
        }
    }
}

// ---------------------------------------------------------------- new_pos (cluster mean)
__global__ void newpos_kernel(const float* __restrict__ pos, float* __restrict__ np) {
    int i = blockIdx.x * blockDim.x + threadIdx.x;
    if (i >= C_CLUST * 3) return;
    int c = i / 3, d = i % 3;
    float s = 0.f;
#pragma unroll
    for (int v = 0; v < PF; ++v) s += pos[(size_t)(c * PF + v) * 3 + d];
    np[i] = s * (1.0f / PF);
}

// ---------------------------------------------------------------- cluster GEMM + reduce
// One wave per 16 nodes (= 4 whole clusters): WMMA feat@W_gather, add
// (pos - new_pos[c])@W_gattr, reduce rows in groups of 4 in-register.
__global__ void gather_kernel(const float* __restrict__ outbuf,
                              const float* __restrict__ pos,
                              const float* __restrict__ np,
                              const float* __restrict__ Wg,
                              const float* __restrict__ Wga,
                              float*       __restrict__ xnew) {
    int wave = blockIdx.x;              // blockDim.x == 32
    int lane = threadIdx.x;
    int row  = lane & 15;
    int half = lane >> 4;
    int n0   = wave * 16;

    const float* fr = outbuf + (size_t)(n0 + row) * DINTER + DBLOOM;  // feat row

    v8f acc[8] = {};
#pragma unroll 1
    for (int kb = 0; kb < DOUT; kb += 4) {
        int k = kb + 2 * half;
        v2f a; a.x = fr[k]; a.y = fr[k + 1];
#pragma unroll
        for (int j = 0; j < 8; ++j) {
            int col = j * 16 + row;
            v2f b; b.x = Wg[(size_t)k * DOUT + col];
                   b.y = Wg[(size_t)(k + 1) * DOUT + col];
            acc[j] = __builtin_amdgcn_wmma_f32_16x16x4_f32(
                false, a, false, b, (short)0, acc[j], false, false);
        }
    }

    float gx[8], gy[8], gz[8];
#pragma unroll
    for (int v = 0; v < 8; ++v) {
        int n = n0 + v + 8 * half;
        int c = n >> 2;
        gx[v] = pos[(size_t)n * 3 + 0] - np[(size_t)c * 3 + 0];
        gy[v] = pos[(size_t)n * 3 + 1] - np[(size_t)c * 3 + 1];
        gz[v] = pos[(size_t)n * 3 + 2] - np[(size_t)c * 3 + 2];
    }

    int cbase = wave * 4 + half * 2;    // lane<16 owns clusters +0,+1 ; lane>=16 owns +2,+3
#pragma unroll
    for (int j = 0; j < 8; ++j) {
        int col = j * 16 + row;
        float w0 = Wga[col], w1 = Wga[DOUT + col], w2 = Wga[2 * DOUT + col];
        float t[8];
#pragma unroll
        for (int v = 0; v < 8; ++v)
            t[v] = acc[j][v] + gx[v] * w0 + gy[v] * w1 + gz[v] * w2;
        float s0 = t[0] + t[1] + t[2] + t[3];
        float s1 = t[4] + t[5] + t[6] + t[7];
        xnew[(size_t)cbase * DOUT + col]       = s0 * (1.0f / PF);
        xnew[(size_t)(cbase + 1) * DOUT + col] = s1 * (1.0f / PF);
    }
}

// ---------------------------------------------------------------- enc keys + sentinel pad
__global__ void enc_kernel(const int* __restrict__ ei, int* __restrict__ sb) {
    int i = blockIdx.x * blockDim.x + threadIdx.x;
    if (i < N_EDGES) {
        int sc = ei[i] / PF;
        int dc = ei[N_EDGES + i] / PF;
        sb[i] = sc * C_CLUST + dc;
    } else if (i < SORT_N) {
        sb[i] = SENTINEL;
    }
}

// ---------------------------------------------------------------- bitonic sort step
__global__ void bitonic_kernel(int* __restrict__ a, int j, int k) {
    unsigned i   = blockIdx.x * blockDim.x + threadIdx.x;
    unsigned ixj = i ^ (unsigned)j;
    if (ixj > i) {
        int ai = a[i], aj = a[ixj];
        bool up = ((i & (unsigned)k) == 0);
        if (up ? (ai > aj) : (ai < aj)) { a[i] = aj; a[ixj] = ai; }
    }
}

// ---------------------------------------------------------------- streaming scan + unique compact
__global__ void compact_kernel(const int* __restrict__ a, int* __restrict__ u,
                               int* __restrict__ cnt) {
    __shared__ int sd[1024];
    __shared__ int carry;
    int tid = threadIdx.x;
    if (tid == 0) carry = 0;
    __syncthreads();
    for (int base = 0; base < N_EDGES; base += 1024) {
        int i = base + tid;
        int keep = 0, val = 0;
        if (i < N_EDGES) {
            val  = a[i];
            keep = (val != SENTINEL) && (i == 0 || a[i - 1] != val);
        }
        sd[tid] = keep;
        __syncthreads();
        for (int off = 1; off < 1024; off <<= 1) {
            int v = (tid >= off) ? sd[tid - off] : 0;
            __syncthreads();
            sd[tid] += v;
            __syncthreads();
        }
        int pos = carry + sd[tid] - keep;
        if (keep) u[pos] = val;
        __syncthreads();
        if (tid == 0) carry += sd[1023];
        __syncthreads();
    }
    if (tid == 0) *cnt = carry;
}

// ---------------------------------------------------------------- new edges out
__global__ void edges_out_kernel(const int* __restrict__ u, const int* __restrict__ cnt,
                                 const float* __restrict__ np,
                                 float* __restrict__ eidx, float* __restrict__ eattr) {
    int i = blockIdx.x * blockDim.x + threadIdx.x;
    if (i >= N_EDGES) return;
    int count = *cnt;
    int v  = (i < count) ? u[i] : 0;   // fill_value = 0
    int sc = v / C_CLUST;
    int dc = v % C_CLUST;
    eidx[i]           = (float)sc;
    eidx[N_EDGES + i] = (float)dc;
    eattr[(size_t)i * 3 + 0] = np[(size_t)dc * 3 + 0] - np[(size_t)sc * 3 + 0];
    eattr[(size_t)i * 3 + 1] = np[(size_t)dc * 3 + 1] - np[(size_t)sc * 3 + 1];
    eattr[(size_t)i * 3 + 2] = np[(size_t)dc * 3 + 2] - np[(size_t)sc * 3 + 2];
}

// ---------------------------------------------------------------- segment max of batch
__global__ void batch_kernel(const int* __restrict__ batch, float* __restrict__ nb) {
    int c = blockIdx.x * blockDim.x + threadIdx.x;
    if (c >= C_CLUST) return;
    int m = batch[c * PF];
#pragma unroll
    for (int v = 1; v < PF; ++v) { int b = batch[c * PF + v]; m = b > m ? b : m; }
    nb[c] = (float)m;
}

// ---------------------------------------------------------------- launcher
extern "C" void kernel_launch(void* const* d_in, const int* in_sizes, int n_in,
                              void* d_out, int out_size, void* d_ws, size_t ws_size,
                              hipStream_t stream) {
    (void)in_sizes; (void)n_in; (void)out_size; (void)ws_size;

    const float* x     = (const float*)d_in[0];
    const float* pos   = (const float*)d_in[1];
    const int*   ei    = (const int*)d_in[2];
    const float* ea    = (const float*)d_in[3];
    const int*   batch = (const int*)d_in[4];
    const float* Wc    = (const float*)d_in[5];
    const float* We    = (const float*)d_in[6];
    /* d_in[7] = D_bloom: dead — bloom_pos never reaches any returned output */
    const float* Wg    = (const float*)d_in[8];
    const float* Wga   = (const float*)d_in[9];

    // output layout: x_new | new_pos | new_edge_index(2,E) | new_edge_attr(E,3) | new_batch
    float* xnew   = (float*)d_out;
    float* np     = xnew + (size_t)C_CLUST * DOUT;
    float* eidx   = np + (size_t)C_CLUST * 3;
    float* eattr  = eidx + (size_t)2 * N_EDGES;
    float* nbatch = eattr + (size_t)N_EDGES * 3;

    // workspace layout
    char* ws      = (char*)d_ws;
    float* outbuf = (float*)ws;                                        // N*144 f32
    int* sortbuf  = (int*)(ws + (size_t)N_NODES * DINTER * 4);         // SORT_N i32
    int* ubuf     = (int*)((char*)sortbuf + (size_t)SORT_N * 4);       // E i32
    int* cnt      = (int*)((char*)ubuf + (size_t)N_EDGES * 4);         // 1 i32

    const int OUTN = N_NODES * DINTER;
    zero_f32_kernel<<<(OUTN + 255) / 256, 256, 0, stream>>>(outbuf, OUTN);

    // 100000 edge tiles, 8 waves/block -> 12500 blocks
    edge_msg_kernel<<<N_EDGES / 16 / 8, 256, 0, stream>>>(x, ei, ea, Wc, We, outbuf);

    newpos_kernel<<<(C_CLUST * 3 + 255) / 256, 256, 0, stream>>>(pos, np);

    // 6250 node tiles, one wave per block
    gather_kernel<<<N_NODES / 16, 32, 0, stream>>>(outbuf, pos, np, Wg, Wga, xnew);

    enc_kernel<<<SORT_N / 256, 256, 0, stream>>>(ei, sortbuf);
    for (int k = 2; k <= SORT_N; k <<= 1)
        for (int j = k >> 1; j > 0; j >>= 1)
            bitonic_kernel<<<SORT_N / 256, 256, 0, stream>>>(sortbuf, j, k);

    compact_kernel<<<1, 1024, 0, stream>>>(sortbuf, ubuf, cnt);
    edges_out_kernel<<<(N_EDGES + 255) / 256, 256, 0, stream>>>(ubuf, cnt, np, eidx, eattr);
    batch_kernel<<<(C_CLUST + 255) / 256, 256, 0, stream>>>(batch, nbatch);
}